// _MambaResidualBlock_22797686407838
// MI455X (gfx1250) — compile-verified
//
#include <hip/hip_runtime.h>
#include <hip/hip_bf16.h>
#include <math.h>

typedef __attribute__((ext_vector_type(2))) float v2f;
typedef __attribute__((ext_vector_type(8))) float v8f;

#define D_MODEL 2048
#define D_INNER 4096
#define D_STATE 16
#define DT_RANK 128
#define BATCH   128
#define EPS     1e-5f

// ---------------------------------------------------------------------------
// Kernel 1: RMSNorm over D_MODEL per batch row. One 256-thread block per row.
// ---------------------------------------------------------------------------
__global__ void __launch_bounds__(256) rmsnorm_kernel(
    const float* __restrict__ hs, const float* __restrict__ w,
    float* __restrict__ out)
{
    const int b = blockIdx.x;
    const float* x = hs + (size_t)b * D_MODEL;
    float s = 0.f;
    for (int i = threadIdx.x; i < D_MODEL; i += 256) {
        float v = x[i];
        s += v * v;
    }
    for (int off = 16; off > 0; off >>= 1)
        s += __shfl_down(s, off, 32);
    __shared__ float red[8];
    const int wid = threadIdx.x >> 5;
    if ((threadIdx.x & 31) == 0) red[wid] = s;
    __syncthreads();
    if (threadIdx.x == 0) {
        float t = 0.f;
        #pragma unroll
        for (int i = 0; i < 8; i++) t += red[i];
        red[0] = rsqrtf(t / (float)D_MODEL + EPS);
    }
    __syncthreads();
    const float r = red[0];
    for (int i = threadIdx.x; i < D_MODEL; i += 256)
        out[(size_t)b * D_MODEL + i] = x[i] * r * w[i];
}

// ---------------------------------------------------------------------------
// Epilogue helper shared by the GEMM kernels.
//   EPI 0: plain store
//   EPI 1: v = softplus(v + bias[col])       (dt path)
//   EPI 2: v = v + add[row*ldc + col]        (residual path)
// ---------------------------------------------------------------------------
template <int EPI>
__device__ __forceinline__ void store_tile(
    const v8f& acc, float* __restrict__ C, int ldc,
    int m0, int n0, int half, int lm,
    const float* __restrict__ bias, const float* __restrict__ add)
{
    const int col = n0 + lm;
    #pragma unroll
    for (int r = 0; r < 8; r++) {
        const int row = m0 + r + half * 8;
        float v = acc[r];
        if (EPI == 1) {
            v += bias[col];
            v = (v > 20.f) ? v : log1pf(__expf(v));   // softplus
        }
        if (EPI == 2) {
            v += add[(size_t)row * ldc + col];
        }
        C[(size_t)row * ldc + col] = v;
    }
}

// ---------------------------------------------------------------------------
// Register-blocked WMMA fp32 GEMM: C[M x N] = X[M x K] @ W[N x K]^T.
// One wave computes a 32x32 output tile: 2 M-tiles x 2 N-tiles, 4 f32
// accumulators (32 VGPRs). Inner step: 4 global_load_b64 -> 4
// v_wmma_f32_16x16x4_f32, i.e. 256 B of cache traffic per WMMA (2x better
// than one-tile-per-wave), while HBM still sees each weight exactly once
// (L2-resident reuse across the 4 M-blocks).
// ISA A-layout for 16x16x4: lanes 0-15 hold K=k,k+1; lanes 16-31 K=k+2,k+3.
// Requires M % 32 == 0 and N % 32 == 0.
// ---------------------------------------------------------------------------
template <int EPI>
__global__ void __launch_bounds__(32) wmma_gemm_blocked(
    const float* __restrict__ X, int lda,
    const float* __restrict__ W, int ldb,
    float* __restrict__ C, int ldc,
    int Kdim,
    const float* __restrict__ bias,
    const float* __restrict__ add)
{
    const int lane = threadIdx.x;           // full wave, EXEC all 1s
    const int half = lane >> 4;
    const int lm   = lane & 15;

    const int n0 = blockIdx.x * 32;
    const int m0 = blockIdx.y * 32;

    const float* ap0 = X + (size_t)(m0 + lm) * lda + half * 2;
    const float* ap1 = ap0 + (size_t)16 * lda;
    const float* bp0 = W + (size_t)(n0 + lm) * ldb + half * 2;
    const float* bp1 = bp0 + (size_t)16 * ldb;

    v8f acc00 = {}, acc01 = {}, acc10 = {}, acc11 = {};

    #pragma unroll 4
    for (int k = 0; k < Kdim; k += 4) {
        v2f a0 = *(const v2f*)(ap0 + k);
        v2f a1 = *(const v2f*)(ap1 + k);
        v2f b0 = *(const v2f*)(bp0 + k);
        v2f b1 = *(const v2f*)(bp1 + k);
        acc00 = __builtin_amdgcn_wmma_f32_16x16x4_f32(false, a0, false, b0,
                                                      (short)0, acc00, false, false);
        acc01 = __builtin_amdgcn_wmma_f32_16x16x4_f32(false, a0, false, b1,
                                                      (short)0, acc01, false, false);
        acc10 = __builtin_amdgcn_wmma_f32_16x16x4_f32(false, a1, false, b0,
                                                      (short)0, acc10, false, false);
        acc11 = __builtin_amdgcn_wmma_f32_16x16x4_f32(false, a1, false, b1,
                                                      (short)0, acc11, false, false);
    }

    store_tile<EPI>(acc00, C, ldc, m0,      n0,      half, lm, bias, add);
    store_tile<EPI>(acc01, C, ldc, m0,      n0 + 16, half, lm, bias, add);
    store_tile<EPI>(acc10, C, ldc, m0 + 16, n0,      half, lm, bias, add);
    store_tile<EPI>(acc11, C, ldc, m0 + 16, n0 + 16, half, lm, bias, add);
}

// ---------------------------------------------------------------------------
// Single-tile WMMA GEMM: one 16x16 tile per wave (x_proj, N=160).
// ---------------------------------------------------------------------------
template <int EPI>
__global__ void __launch_bounds__(32) wmma_gemm_kernel(
    const float* __restrict__ X, int lda,
    const float* __restrict__ W, int ldb,
    float* __restrict__ C, int ldc,
    int Kdim,
    const float* __restrict__ bias,
    const float* __restrict__ add)
{
    const int lane = threadIdx.x;
    const int half = lane >> 4;
    const int lm   = lane & 15;

    const int n0 = blockIdx.x * 16;
    const int m0 = blockIdx.y * 16;

    const float* ap = X + (size_t)(m0 + lm) * lda + half * 2;
    const float* bp = W + (size_t)(n0 + lm) * ldb + half * 2;

    v8f acc = {};
    #pragma unroll 8
    for (int k = 0; k < Kdim; k += 4) {
        v2f a = *(const v2f*)(ap + k);
        v2f b = *(const v2f*)(bp + k);
        acc = __builtin_amdgcn_wmma_f32_16x16x4_f32(false, a, false, b,
                                                    (short)0, acc, false, false);
    }
    store_tile<EPI>(acc, C, ldc, m0, n0, half, lm, bias, add);
}

// ---------------------------------------------------------------------------
// Kernel 3: depthwise conv (K=4) + bias + SiLU; emits new_conv_state.
// ---------------------------------------------------------------------------
__global__ void __launch_bounds__(256) conv_silu_kernel(
    const float* __restrict__ xz,          // [B, 2*D_INNER], xc = [:, :D_INNER]
    const float* __restrict__ conv_state,  // [B, D_INNER, 3]
    const float* __restrict__ conv_w,      // [D_INNER, 4]
    const float* __restrict__ conv_b,      // [D_INNER]
    float* __restrict__ xact,              // [B, D_INNER]
    float* __restrict__ new_conv_state)    // [B, D_INNER, 3]
{
    const int idx = blockIdx.x * 256 + threadIdx.x;   // B*D_INNER threads
    const int b = idx >> 12;
    const int d = idx & (D_INNER - 1);

    const float* cs = conv_state + (size_t)idx * 3;
    const float c0 = cs[0], c1 = cs[1], c2 = cs[2];
    const float x3 = xz[(size_t)b * (2 * D_INNER) + d];

    const float* wv = conv_w + (size_t)d * 4;
    float s = c0 * wv[0] + c1 * wv[1] + c2 * wv[2] + x3 * wv[3] + conv_b[d];
    const float act = s / (1.f + __expf(-s));         // SiLU
    xact[idx] = act;

    float* ncs = new_conv_state + (size_t)idx * 3;
    ncs[0] = c1; ncs[1] = c2; ncs[2] = x3;
}

// ---------------------------------------------------------------------------
// Kernel 6: SSM recurrence over the 16 states + D skip + SiLU(z) gate.
// ---------------------------------------------------------------------------
__global__ void __launch_bounds__(256) ssm_kernel(
    const float* __restrict__ dtbuf,       // [B, D_INNER] (post-softplus)
    const float* __restrict__ proj,        // [B, 160] -> Bmat @128, Cmat @144
    const float* __restrict__ A_log,       // [D_INNER, 16]
    const float* __restrict__ Dv,          // [D_INNER]
    const float* __restrict__ xact,        // [B, D_INNER]
    const float* __restrict__ ssm_state,   // [B, D_INNER, 16]
    const float* __restrict__ xz,          // [B, 2*D_INNER], z = [:, D_INNER:]
    float* __restrict__ ybuf,              // [B, D_INNER]
    float* __restrict__ new_ssm)           // [B, D_INNER, 16]
{
    const int idx = blockIdx.x * 256 + threadIdx.x;   // B*D_INNER threads
    const int b = idx >> 12;
    const int d = idx & (D_INNER - 1);

    const float dt = dtbuf[idx];
    const float xa = xact[idx];
    const float* st = ssm_state + (size_t)idx * D_STATE;
    const float* al = A_log + (size_t)d * D_STATE;
    const float* Bm = proj + (size_t)b * 160 + DT_RANK;
    const float* Cm = Bm + D_STATE;
    float* ns = new_ssm + (size_t)idx * D_STATE;

    float y = 0.f;
    #pragma unroll
    for (int s = 0; s < D_STATE; s++) {
        const float dA = __expf(-__expf(al[s]) * dt);
        const float v = st[s] * dA + dt * Bm[s] * xa;
        ns[s] = v;
        y += v * Cm[s];
    }
    y += Dv[d] * xa;

    const float z = xz[(size_t)b * (2 * D_INNER) + D_INNER + d];
    y *= z / (1.f + __expf(-z));                      // * SiLU(z)
    ybuf[idx] = y;
}

// ---------------------------------------------------------------------------
// Host launcher
// ---------------------------------------------------------------------------
extern "C" void kernel_launch(void* const* d_in, const int* in_sizes, int n_in,
                              void* d_out, int out_size, void* d_ws, size_t ws_size,
                              hipStream_t stream)
{
    const float* hidden    = (const float*)d_in[0];   // [128,1,2048]
    const float* convstate = (const float*)d_in[1];   // [128,4096,3]
    const float* ssmstate  = (const float*)d_in[2];   // [128,4096,16]
    const float* norm_w    = (const float*)d_in[3];   // [2048]
    const float* in_proj_w = (const float*)d_in[4];   // [8192,2048]
    const float* conv_w    = (const float*)d_in[5];   // [4096,4]
    const float* conv_b    = (const float*)d_in[6];   // [4096]
    const float* x_proj_w  = (const float*)d_in[7];   // [160,4096]
    const float* dt_proj_w = (const float*)d_in[8];   // [4096,128]
    const float* dt_proj_b = (const float*)d_in[9];   // [4096]
    const float* A_log     = (const float*)d_in[10];  // [4096,16]
    const float* Dvec      = (const float*)d_in[11];  // [4096]
    const float* out_proj_w= (const float*)d_in[12];  // [2048,4096]

    float* out = (float*)d_out;
    float* out_output   = out;                           // 128*2048
    float* out_convstat = out + 128 * 2048;              // 128*4096*3
    float* out_ssmstat  = out_convstat + 128 * 4096 * 3; // 128*4096*16

    float* ws = (float*)d_ws;
    float* normed = ws;                      size_t off = (size_t)BATCH * D_MODEL;
    float* xz     = ws + off;                off += (size_t)BATCH * 2 * D_INNER;
    float* xact   = ws + off;                off += (size_t)BATCH * D_INNER;
    float* proj   = ws + off;                off += (size_t)BATCH * 160;
    float* dtbuf  = ws + off;                off += (size_t)BATCH * D_INNER;
    float* ybuf   = ws + off;

    // 1) RMSNorm
    rmsnorm_kernel<<<BATCH, 256, 0, stream>>>(hidden, norm_w, normed);

    // 2) xz = normed @ in_proj_w^T   (M=128, N=8192, K=2048)
    wmma_gemm_blocked<0><<<dim3((2 * D_INNER) / 32, BATCH / 32), 32, 0, stream>>>(
        normed, D_MODEL, in_proj_w, D_MODEL, xz, 2 * D_INNER, D_MODEL,
        nullptr, nullptr);

    // 3) conv + SiLU, emit new_conv_state
    conv_silu_kernel<<<(BATCH * D_INNER) / 256, 256, 0, stream>>>(
        xz, convstate, conv_w, conv_b, xact, out_convstat);

    // 4) proj = xact @ x_proj_w^T    (M=128, N=160, K=4096)
    wmma_gemm_kernel<0><<<dim3(160 / 16, BATCH / 16), 32, 0, stream>>>(
        xact, D_INNER, x_proj_w, D_INNER, proj, 160, D_INNER,
        nullptr, nullptr);

    // 5) dt = softplus(dt_raw @ dt_proj_w^T + b)  (M=128, N=4096, K=128)
    //    dt_raw = proj[:, :128] with row stride 160
    wmma_gemm_blocked<1><<<dim3(D_INNER / 32, BATCH / 32), 32, 0, stream>>>(
        proj, 160, dt_proj_w, DT_RANK, dtbuf, D_INNER, DT_RANK,
        dt_proj_b, nullptr);

    // 6) SSM update + gate, emit new_ssm_state
    ssm_kernel<<<(BATCH * D_INNER) / 256, 256, 0, stream>>>(
        dtbuf, proj, A_log, Dvec, xact, ssmstate, xz, ybuf, out_ssmstat);

    // 7) output = y @ out_proj_w^T + hidden   (M=128, N=2048, K=4096)
    wmma_gemm_blocked<2><<<dim3(D_MODEL / 32, BATCH / 32), 32, 0, stream>>>(
        ybuf, D_INNER, out_proj_w, D_INNER, out_output, D_MODEL, D_INNER,
        nullptr, hidden);

    (void)in_sizes; (void)n_in; (void)out_size; (void)ws_size;
}